// NGCF_13099650253234
// MI455X (gfx1250) — compile-verified
//
#include <hip/hip_runtime.h>
#include <hip/hip_bf16.h>
#include <math.h>

#define N_USER  100000
#define N_ITEM  50000
#define NTOT    150000
#define E_EDGES 2400000
#define D       64
#define BATCH   4096
#define SLOPE   0.2f

typedef __attribute__((ext_vector_type(2))) float v2f;
typedef __attribute__((ext_vector_type(8))) float v8f;

// ---------------------------------------------------------------------------
// ego = concat(user_emb, item_emb); allEmb[:, 0:64] = ego (layer-0, raw)
// ---------------------------------------------------------------------------
__global__ void init_kernel(const float* __restrict__ uemb,
                            const float* __restrict__ iemb,
                            float* __restrict__ ego,
                            float* __restrict__ allEmb) {
    int idx = blockIdx.x * blockDim.x + threadIdx.x;
    if (idx >= NTOT * D) return;
    float v = (idx < N_USER * D) ? uemb[idx] : iemb[idx - N_USER * D];
    ego[idx] = v;
    int row = idx >> 6, col = idx & 63;
    allEmb[row * 256 + col] = v;
}

__global__ void zero_kernel(float* __restrict__ p, int n) {
    int idx = blockIdx.x * blockDim.x + threadIdx.x;
    if (idx < n) p[idx] = 0.f;
}

// ---------------------------------------------------------------------------
// SpMM scatter: side[rows[e], :] += vals[e] * ego[cols[e], :]
// 16 threads per edge, float4 gather (L2-resident table), 4 f32 atomics each.
// ---------------------------------------------------------------------------
__global__ void spmm_scatter(const int* __restrict__ rows,
                             const int* __restrict__ cols,
                             const float* __restrict__ vals,
                             const float* __restrict__ ego,
                             float* __restrict__ side) {
    long tid = (long)blockIdx.x * blockDim.x + threadIdx.x;
    if (tid >= (long)E_EDGES * 16) return;
    int e = (int)(tid >> 4);
    int q = (int)(tid & 15);
    int r = rows[e], c = cols[e];
    float v = vals[e];
    const float4 x = *reinterpret_cast<const float4*>(&ego[c * D + q * 4]);
    float* dst = &side[r * D + q * 4];
    unsafeAtomicAdd(dst + 0, v * x.x);
    unsafeAtomicAdd(dst + 1, v * x.y);
    unsafeAtomicAdd(dst + 2, v * x.z);
    unsafeAtomicAdd(dst + 3, v * x.w);
}

// ---------------------------------------------------------------------------
// Fused: sum = side@Wgc + bgc ; bi = (ego*side)@Wbi + bbi ;
//        ego_new = leaky_relu(sum+bi) ; slab cols = l2norm(ego_new)
// Block = 128 threads (4 waves). Wave w owns output column tile w (16 cols).
// Each wave: 16 K-steps of V_WMMA_F32_16X16X4_F32, two accumulators.
// A 16x4 layout: lanes0-15 -> K=0,1 ; lanes16-31 -> K=2,3 (per lane VGPR0/1).
// B 4x16 layout: VGPR0 = rows {K0 | K1}, VGPR1 = rows {K2 | K3} across halves.
// C/D 16x16:     VGPR v -> M = v + 8*(lane>=16), N = lane&15.
// ---------------------------------------------------------------------------
__global__ __launch_bounds__(128)
void gemm_fused(const float* __restrict__ side,
                float* __restrict__ ego,          // read (bi path), then overwritten
                const float* __restrict__ Wgc, const float* __restrict__ bgc,
                const float* __restrict__ Wbi, const float* __restrict__ bbi,
                float* __restrict__ allEmbCol) {  // allEmb + (k+1)*64
    __shared__ float tile[16 * D];
    __shared__ float part[16 * 8];
    __shared__ float rnorm[16];

    const int tid  = threadIdx.x;
    const int wave = tid >> 5;       // output column tile 0..3
    const int lane = tid & 31;
    const int l15  = lane & 15;
    const int hi   = lane >> 4;      // 0: lanes 0-15, 1: lanes 16-31

    const int rowBase = blockIdx.x * 16;
    const float* aRow = side + (rowBase + l15) * D;
    const float* eRow = ego  + (rowBase + l15) * D;
    const int colBase = wave * 16 + l15;

    v8f cgc = {};
    v8f cbi = {};
#pragma unroll
    for (int kk = 0; kk < 16; ++kk) {
        const int ka = kk * 4 + 2 * hi;          // A fragment K base for this half
        v2f aS, eF, bG, bB;
        aS.x = aRow[ka];     aS.y = aRow[ka + 1];
        eF.x = eRow[ka];     eF.y = eRow[ka + 1];
        v2f aB = aS * eF;                         // (ego * side) fragment
        const int kb0 = kk * 4 + hi;              // B fragment rows
        const int kb1 = kk * 4 + 2 + hi;
        bG.x = Wgc[kb0 * D + colBase]; bG.y = Wgc[kb1 * D + colBase];
        bB.x = Wbi[kb0 * D + colBase]; bB.y = Wbi[kb1 * D + colBase];
        cgc = __builtin_amdgcn_wmma_f32_16x16x4_f32(false, aS, false, bG,
                                                    (short)0, cgc, false, false);
        cbi = __builtin_amdgcn_wmma_f32_16x16x4_f32(false, aB, false, bB,
                                                    (short)0, cbi, false, false);
    }

    // All intra-block reads of ego rows are done; safe to overwrite in place.
    __syncthreads();

    const int   col   = wave * 16 + l15;
    const float biasG = bgc[col];
    const float biasB = bbi[col];
#pragma unroll
    for (int v = 0; v < 8; ++v) {
        const int M = v + 8 * hi;
        float val = cgc[v] + biasG + cbi[v] + biasB;
        float act = (val >= 0.f) ? val : SLOPE * val;
        ego[(rowBase + M) * D + col] = act;       // un-normalized, feeds next layer
        tile[M * D + col] = act;
    }
    __syncthreads();

    // Row L2 norms over all 64 columns (16 rows x 8 partials)
    {
        const int r = tid >> 3, cs = (tid & 7) * 8;
        float s = 0.f;
#pragma unroll
        for (int j = 0; j < 8; ++j) { float t = tile[r * D + cs + j]; s += t * t; }
        part[r * 8 + (tid & 7)] = s;
    }
    __syncthreads();
    if (tid < 16) {
        float s = 0.f;
#pragma unroll
        for (int j = 0; j < 8; ++j) s += part[tid * 8 + j];
        rnorm[tid] = 1.0f / fmaxf(sqrtf(s), 1e-12f);
    }
    __syncthreads();
    {
        const int r = tid >> 3, cs = (tid & 7) * 8;
        const float inv = rnorm[r];
#pragma unroll
        for (int j = 0; j < 8; ++j)
            allEmbCol[(rowBase + r) * 256 + cs + j] = tile[r * D + cs + j] * inv;
    }
}

// ---------------------------------------------------------------------------
// Gather u_g / pos_g / neg_g rows (256 floats each) into d_out, float4 copies.
// ---------------------------------------------------------------------------
__global__ void gather_kernel(const int* __restrict__ users,
                              const int* __restrict__ pos,
                              const int* __restrict__ neg,
                              const float* __restrict__ allEmb,
                              float* __restrict__ out) {
    int idx = blockIdx.x * blockDim.x + threadIdx.x;   // 3*BATCH*64 float4 slots
    if (idx >= 3 * BATCH * 64) return;
    int rowsel = idx >> 6;
    int c4 = (idx & 63) * 4;
    int which = rowsel / BATCH;
    int i = rowsel - which * BATCH;
    int r = (which == 0) ? users[i]
          : (which == 1) ? N_USER + pos[i]
                         : N_USER + neg[i];
    float4 v = *reinterpret_cast<const float4*>(&allEmb[r * 256 + c4]);
    *reinterpret_cast<float4*>(&out[rowsel * 256 + c4]) = v;
}

// ---------------------------------------------------------------------------
extern "C" void kernel_launch(void* const* d_in, const int* in_sizes, int n_in,
                              void* d_out, int out_size, void* d_ws, size_t ws_size,
                              hipStream_t stream) {
    const int*   users = (const int*)d_in[0];
    const int*   pos   = (const int*)d_in[1];
    const int*   neg   = (const int*)d_in[2];
    const int*   arows = (const int*)d_in[3];
    const int*   acols = (const int*)d_in[4];
    const float* avals = (const float*)d_in[5];
    const float* uemb  = (const float*)d_in[6];
    const float* iemb  = (const float*)d_in[7];
    // dict order per layer: W_gc_k, b_gc_k, W_bi_k, b_bi_k
    const float* Wgc[3] = {(const float*)d_in[8],  (const float*)d_in[12], (const float*)d_in[16]};
    const float* bgc[3] = {(const float*)d_in[9],  (const float*)d_in[13], (const float*)d_in[17]};
    const float* Wbi[3] = {(const float*)d_in[10], (const float*)d_in[14], (const float*)d_in[18]};
    const float* bbi[3] = {(const float*)d_in[11], (const float*)d_in[15], (const float*)d_in[19]};

    float* allEmb = (float*)d_ws;                        // NTOT*256 floats (153.6 MB)
    float* ego    = allEmb + (size_t)NTOT * 256;         // NTOT*64  floats (38.4 MB)
    float* side   = ego    + (size_t)NTOT * 64;          // NTOT*64  floats (38.4 MB)

    init_kernel<<<(NTOT * D + 255) / 256, 256, 0, stream>>>(uemb, iemb, ego, allEmb);

    for (int k = 0; k < 3; ++k) {
        zero_kernel<<<(NTOT * D + 255) / 256, 256, 0, stream>>>(side, NTOT * D);
        long nthreads = (long)E_EDGES * 16;
        spmm_scatter<<<(int)((nthreads + 255) / 256), 256, 0, stream>>>(
            arows, acols, avals, ego, side);
        gemm_fused<<<NTOT / 16, 128, 0, stream>>>(
            side, ego, Wgc[k], bgc[k], Wbi[k], bbi[k], allEmb + (size_t)(k + 1) * 64);
    }

    gather_kernel<<<(3 * BATCH * 64 + 255) / 256, 256, 0, stream>>>(
        users, pos, neg, allEmb, (float*)d_out);
}